// MediumLSTM_66563403153603
// MI455X (gfx1250) — compile-verified
//
#include <hip/hip_runtime.h>
#include <hip/hip_bf16.h>

typedef float v2f __attribute__((ext_vector_type(2)));
typedef float v4f __attribute__((ext_vector_type(4)));
typedef float v8f __attribute__((ext_vector_type(8)));

#define IN    128
#define TSTEPS 64
#define BATCH 8192
#define H1    32
#define H2    16
#define H3    8

// Concatenated K per layer: [x | h]
#define K1 (IN + H1)   // 160
#define K2 (H1 + H2)   // 48
#define K3 (H2 + H3)   // 24
// LDS row strides:
//  SW1 = 164 (mult. of 4 -> 16B-aligned float4 x-staging; 164%64=36=4*9 ->
//        16 rows hit 16 distinct bank groups, and s*dm= -2 mod 64 unsolvable)
//  SW2 = 50, SW3 = 26 (even -> 8B-aligned b64 frags; residues 50/26 give
//        conflict-free half-wave patterns, cross-half collision at |dm|>15)
#define SW1 164
#define SW2 50
#define SW3 26

#define W1_FL (128 * SW1)
#define W2_FL (64 * SW2)
#define W3_FL (32 * SW3)
#define XH_FL (16 * (SW1 + SW2 + SW3))   // per-wave staging (3840 floats)
#define WAVES 2
#define SMEM_FL (W1_FL + W2_FL + W3_FL + 128 + 64 + 32 + WAVES * XH_FL)
// = 32928 floats = 131,712 B  ->  two workgroups co-resident per 320KB WGP

__device__ __forceinline__ float fsig(float x) {
    return 1.0f / (1.0f + __expf(-x));
}
__device__ __forceinline__ float ftanh(float x) {
    x = fminf(fmaxf(x, -15.0f), 15.0f);
    float e = __expf(2.0f * x);
    return (e - 1.0f) / (e + 1.0f);
}

__device__ __forceinline__ v8f wmma4(v2f a, v2f b, v8f c) {
    // D = A(16x4 f32) * B(4x16 f32) + C(16x16 f32)
    return __builtin_amdgcn_wmma_f32_16x16x4_f32(
        /*neg_a=*/false, a, /*neg_b=*/false, b,
        /*c_mod=*/(short)0, c, /*reuse_a=*/false, /*reuse_b=*/false);
}

// gates[16 x NT*16] = xh[16 x K] * Wcat^T, acc pre-initialized with bias.
template <int NT>
__device__ __forceinline__ void gemm_tiles(const float* __restrict__ xh, int sx,
                                           const float* __restrict__ W, int sw,
                                           const float* __restrict__ bias,
                                           int K, int lane, v8f* acc) {
    const int nlo   = lane & 15;
    const int khalf = (lane >> 4) << 1;  // lanes 0-15 -> K+0/1, lanes 16-31 -> K+2/3
#pragma unroll
    for (int nt = 0; nt < NT; ++nt) {
        float bv = bias[nt * 16 + nlo];
        acc[nt] = (v8f){bv, bv, bv, bv, bv, bv, bv, bv};
    }
    const float* arow = xh + nlo * sx + khalf;
    for (int k0 = 0; k0 < K; k0 += 4) {
        v2f a;
        a.x = arow[k0];
        a.y = arow[k0 + 1];
#pragma unroll
        for (int nt = 0; nt < NT; ++nt) {
            const float* brow = W + (nt * 16 + nlo) * sw + khalf + k0;
            v2f b;
            b.x = brow[0];
            b.y = brow[1];
            acc[nt] = wmma4(a, b, acc[nt]);
        }
    }
}

// LSTM pointwise epilogue for H = HT*16 (layers 1 and 2). Writes h to the
// layer's own h-section and the next layer's x-section.
template <int HT>
__device__ __forceinline__ void lstm_epilogue(const v8f* acc, v8f* c,
                                              float* __restrict__ hdst0, int s0,
                                              float* __restrict__ hdst1, int s1,
                                              int lane) {
    const int nlo  = lane & 15;
    const int mhi8 = (lane >> 4) << 3;
#pragma unroll
    for (int j = 0; j < HT; ++j) {
#pragma unroll
        for (int r = 0; r < 8; ++r) {
            float iv = fsig(acc[0 * HT + j][r]);
            float fv = fsig(acc[1 * HT + j][r]);
            float gv = ftanh(acc[2 * HT + j][r]);
            float ov = fsig(acc[3 * HT + j][r]);
            float cv = fv * c[j][r] + iv * gv;
            c[j][r]  = cv;
            float hv = ov * ftanh(cv);
            int m = r + mhi8;
            int n = j * 16 + nlo;
            hdst0[m * s0 + n] = hv;
            hdst1[m * s1 + n] = hv;
        }
    }
}

__global__ void __launch_bounds__(32 * WAVES, 1)
lstm3_fused_wmma(const float* __restrict__ x,
                 const float* __restrict__ w_ih1, const float* __restrict__ w_hh1,
                 const float* __restrict__ b_ih1, const float* __restrict__ b_hh1,
                 const float* __restrict__ w_ih2, const float* __restrict__ w_hh2,
                 const float* __restrict__ b_ih2, const float* __restrict__ b_hh2,
                 const float* __restrict__ w_ih3, const float* __restrict__ w_hh3,
                 const float* __restrict__ b_ih3, const float* __restrict__ b_hh3,
                 const float* __restrict__ fc_w, const float* __restrict__ fc_b,
                 float* __restrict__ out) {
    extern __shared__ float smem[];
    float* W1 = smem;
    float* W2 = W1 + W1_FL;
    float* W3 = W2 + W2_FL;
    float* B1 = W3 + W3_FL;
    float* B2 = B1 + 128;
    float* B3 = B2 + 64;
    float* XH = B3 + 32;

    const int tid = threadIdx.x;

    // ---- one-time cooperative preload: concat [W_ih | W_hh] per layer ----
    for (int i = tid; i < 128 * K1; i += blockDim.x) {
        int n = i / K1, k = i - n * K1;
        W1[n * SW1 + k] = (k < IN) ? w_ih1[n * IN + k] : w_hh1[n * H1 + (k - IN)];
    }
    for (int i = tid; i < 64 * K2; i += blockDim.x) {
        int n = i / K2, k = i - n * K2;
        W2[n * SW2 + k] = (k < H1) ? w_ih2[n * H1 + k] : w_hh2[n * H2 + (k - H1)];
    }
    for (int i = tid; i < 32 * K3; i += blockDim.x) {
        int n = i / K3, k = i - n * K3;
        W3[n * SW3 + k] = (k < H2) ? w_ih3[n * H2 + k] : w_hh3[n * H3 + (k - H2)];
    }
    for (int i = tid; i < 128; i += blockDim.x) B1[i] = b_ih1[i] + b_hh1[i];
    for (int i = tid; i < 64; i += blockDim.x)  B2[i] = b_ih2[i] + b_hh2[i];
    for (int i = tid; i < 32; i += blockDim.x)  B3[i] = b_ih3[i] + b_hh3[i];
    __syncthreads();

    const int wave = tid >> 5;
    const int lane = tid & 31;
    float* xh1 = XH + wave * XH_FL;        // [16 x SW1] : [x_t | h1]
    float* xh2 = xh1 + 16 * SW1;           // [16 x SW2] : [h1  | h2]
    float* xh3 = xh2 + 16 * SW2;           // [16 x SW3] : [h2  | h3]
    const int rbase = (blockIdx.x * WAVES + wave) * 16;

    // zero initial h-state sections (wave-private, no barrier needed)
    for (int i = lane; i < 16 * H1; i += 32) {
        int m = i >> 5, n = i & 31;
        xh1[m * SW1 + IN + n] = 0.0f;
    }
    for (int i = lane; i < 16 * H2; i += 32) {
        int m = i >> 4, n = i & 15;
        xh2[m * SW2 + H1 + n] = 0.0f;
    }
    for (int i = lane; i < 16 * H3; i += 32) {
        int m = i >> 3, n = i & 7;
        xh3[m * SW3 + H2 + n] = 0.0f;
    }

    const v8f vz = {0.f, 0.f, 0.f, 0.f, 0.f, 0.f, 0.f, 0.f};
    v8f c1[2] = {vz, vz};
    v8f c2[1] = {vz};
    v8f c3    = vz;

    const int nlo  = lane & 15;
    const int mhi8 = (lane >> 4) << 3;

    for (int t = 0; t < TSTEPS; ++t) {
        // ---- stage x_t tile [16 x 128] into LDS (coalesced, streamed) ----
#pragma unroll 4
        for (int m = 0; m < 16; ++m) {
            const float* src = x + ((size_t)(rbase + m) * TSTEPS + t) * IN;
            v4f v = __builtin_nontemporal_load((const v4f*)src + lane);
            *(v4f*)&xh1[m * SW1 + 4 * lane] = v;
            if (t + 1 < TSTEPS)
                __builtin_prefetch(src + IN + 4 * lane, 0, 1);  // next timestep
        }

        // ---- layer 1: gates[16x128] = [x|h1] (K=160) * W1cat^T ----
        v8f g1[8];
        gemm_tiles<8>(xh1, SW1, W1, SW1, B1, K1, lane, g1);
        lstm_epilogue<2>(g1, c1, xh1 + IN, SW1, xh2, SW2, lane);

        // ---- layer 2: gates[16x64] = [h1|h2] (K=48) * W2cat^T ----
        v8f g2[4];
        gemm_tiles<4>(xh2, SW2, W2, SW2, B2, K2, lane, g2);
        lstm_epilogue<1>(g2, c2, xh2 + H1, SW2, xh3, SW3, lane);

        // ---- layer 3: gates[16x32] = [h2|h3] (K=24) * W3cat^T ----
        // H=8: i/f share tile0 (cols 0-7 / 8-15), g/o share tile1.
        v8f g3[2];
        gemm_tiles<2>(xh3, SW3, W3, SW3, B3, K3, lane, g3);
#pragma unroll
        for (int r = 0; r < 8; ++r) {
            float t0 = g3[0][r], t1 = g3[1][r];
            float fpre = __shfl_xor(t0, 8, 32);
            float opre = __shfl_xor(t1, 8, 32);
            float iv = fsig(t0), fv = fsig(fpre);
            float gv = ftanh(t1), ov = fsig(opre);
            float cv = fv * c3[r] + iv * gv;
            c3[r]    = cv;
            float hv = ov * ftanh(cv);
            if (nlo < 8) xh3[(r + mhi8) * SW3 + H2 + nlo] = hv;
        }
    }

    // ---- final FC on h3 at t = T-1 : out[m] = h3[m,:] . fc_w + fc_b ----
    if (lane < 16) {
        float acc = fc_b[0];
#pragma unroll
        for (int k = 0; k < H3; ++k)
            acc += xh3[lane * SW3 + H2 + k] * fc_w[k];
        out[rbase + lane] = acc;
    }
}

extern "C" void kernel_launch(void* const* d_in, const int* in_sizes, int n_in,
                              void* d_out, int out_size, void* d_ws, size_t ws_size,
                              hipStream_t stream) {
    (void)in_sizes; (void)n_in; (void)d_ws; (void)ws_size; (void)out_size;
    const float* x     = (const float*)d_in[0];
    const float* w_ih1 = (const float*)d_in[1];
    const float* w_hh1 = (const float*)d_in[2];
    const float* b_ih1 = (const float*)d_in[3];
    const float* b_hh1 = (const float*)d_in[4];
    const float* w_ih2 = (const float*)d_in[5];
    const float* w_hh2 = (const float*)d_in[6];
    const float* b_ih2 = (const float*)d_in[7];
    const float* b_hh2 = (const float*)d_in[8];
    const float* w_ih3 = (const float*)d_in[9];
    const float* w_hh3 = (const float*)d_in[10];
    const float* b_ih3 = (const float*)d_in[11];
    const float* b_hh3 = (const float*)d_in[12];
    const float* fc_w  = (const float*)d_in[13];
    const float* fc_b  = (const float*)d_in[14];
    float* out = (float*)d_out;

    const size_t smem_bytes = (size_t)SMEM_FL * sizeof(float);
    hipFuncSetAttribute((const void*)lstm3_fused_wmma,
                        hipFuncAttributeMaxDynamicSharedMemorySize,
                        (int)smem_bytes);

    dim3 grid(BATCH / (WAVES * 16));   // 256 workgroups
    dim3 block(32 * WAVES);            // 2 wave32s
    hipLaunchKernelGGL(lstm3_fused_wmma, grid, block, smem_bytes, stream,
                       x, w_ih1, w_hh1, b_ih1, b_hh1,
                       w_ih2, w_hh2, b_ih2, b_hh2,
                       w_ih3, w_hh3, b_ih3, b_hh3,
                       fc_w, fc_b, out);
}